// RNN_36369783062895
// MI455X (gfx1250) — compile-verified
//
#include <hip/hip_runtime.h>
#include <hip/hip_bf16.h>

typedef __attribute__((ext_vector_type(2))) float v2f;
typedef __attribute__((ext_vector_type(8))) float v8f;

#define T_STEPS 256
#define BATCH   256
#define NINP    128
#define NREC    512
#define NOUT    64
#define LDS_STRIDE 516   // even (8B-aligned float2 LDS loads), != 512 to spread banks

#define WMMA_F32(A, B, C) \
    __builtin_amdgcn_wmma_f32_16x16x4_f32(false, (A), false, (B), (short)0, (C), false, false)

// -----------------------------------------------------------------------------
// Fused RNN scan. Grid: 16 blocks (one 16-row batch tile each), 256 threads =
// 8 waves; wave w owns N-columns [w*64, w*64+64) as four 16x16 WMMA tiles.
//
// The two GEMM loops are explicitly software-pipelined in groups of 4 k-steps
// (4 A-fragments + 16 B-fragments + 16 WMMAs per group), double-buffered in
// named registers so ~20 loads are always in flight ahead of the WMMAs that
// consume them (the compiler's scheduler was serializing load->wait->wmma).
// -----------------------------------------------------------------------------
__global__ __launch_bounds__(256, 1) void rnn_scan_kernel(
    const float* __restrict__ inputs,   // [T,B,NINP]
    const float* __restrict__ noise,    // [T,B,NREC]
    const float* __restrict__ Wi,       // [NREC,NINP]
    const float* __restrict__ bi,       // [NREC]
    const float* __restrict__ Wrec,     // [NREC,NREC]
    float* __restrict__ h_out,          // [T,B,NREC]
    float* __restrict__ h_final)        // [B,NREC]
{
    __shared__ float hl[16 * LDS_STRIDE];

    const int tid  = threadIdx.x;
    const int wave = tid >> 5;
    const int lane = tid & 31;
    const int l16  = lane & 15;
    const int lhi  = lane >> 4;          // 0 or 1
    const int b0   = blockIdx.x * 16;    // batch-tile origin
    const int nbase = wave * 64;

    // zero-init hidden tile in LDS (h0 = 0)
    for (int i = tid; i < 16 * LDS_STRIDE; i += 256) hl[i] = 0.0f;

    // per-tile fixed output column (per lane) + bias
    int   ncol[4];
    float bias[4];
#pragma unroll
    for (int tt = 0; tt < 4; ++tt) {
        ncol[tt] = nbase + tt * 16 + l16;
        bias[tt] = bi[ncol[tt]];
    }

    // per-lane invariant fragment base pointers
    const float* arow_in  = inputs + (size_t)(b0 + l16) * NINP + 2 * lhi; // + t*B*NINP + k
    const float* hrow_lds = hl + l16 * LDS_STRIDE + 2 * lhi;              // + k
    const float* brow_wi[4];
    const float* brow_wr[4];
#pragma unroll
    for (int tt = 0; tt < 4; ++tt) {
        brow_wi[tt] = Wi   + (size_t)ncol[tt] * NINP + 2 * lhi;
        brow_wr[tt] = Wrec + (size_t)ncol[tt] * NREC + 2 * lhi;
    }

    // per-lane prefetch bases (single-use HBM streams):
    //  - inputs: 32 lanes x 64 floats (256B) covers the full 16x128 tile
    //  - noise : lane -> (row l16, cacheline lhi) covers the wave's 16x64 slab
    const float* pf_in = inputs + (size_t)b0 * NINP + lane * 64;
    const float* pf_nz = noise + (size_t)(b0 + l16) * NREC + nbase + lhi * 32;

#pragma unroll 1
    for (int t = 0; t < T_STEPS; ++t) {
        // ---- 1) prefetch this step's HBM streams into cache (no VGPR cost) ----
        __builtin_prefetch(pf_in + (size_t)t * BATCH * NINP, 0, 1);
        __builtin_prefetch(pf_nz + (size_t)t * BATCH * NREC, 0, 1);

        __syncthreads();   // hl writes from previous step visible

        v8f acc[4];
#pragma unroll
        for (int tt = 0; tt < 4; ++tt) {
            v8f c;
#pragma unroll
            for (int r = 0; r < 8; ++r) c[r] = bias[tt];
            acc[tt] = c;
        }

        // ---- 2) recurrent: K = NREC, A from LDS, B from L2-resident Wrec ----
        {
            v2f a0[4], a1[4], b0f[4][4], b1f[4][4];
            // prologue: group 0
#pragma unroll
            for (int ks = 0; ks < 4; ++ks) {
                a0[ks] = *(const v2f*)(hrow_lds + 4 * ks);
#pragma unroll
                for (int tt = 0; tt < 4; ++tt)
                    b0f[ks][tt] = *(const v2f*)(brow_wr[tt] + 4 * ks);
            }
#pragma unroll 1
            for (int g = 0; g < NREC / 16; g += 2) {
                const int k1 = (g + 1) * 16;
                const int k2 = (g + 2 < NREC / 16) ? (g + 2) * 16 : 0; // tail: dead reload
                // issue group g+1 loads
#pragma unroll
                for (int ks = 0; ks < 4; ++ks) {
                    a1[ks] = *(const v2f*)(hrow_lds + k1 + 4 * ks);
#pragma unroll
                    for (int tt = 0; tt < 4; ++tt)
                        b1f[ks][tt] = *(const v2f*)(brow_wr[tt] + k1 + 4 * ks);
                }
                // consume group g
#pragma unroll
                for (int ks = 0; ks < 4; ++ks)
#pragma unroll
                    for (int tt = 0; tt < 4; ++tt)
                        acc[tt] = WMMA_F32(a0[ks], b0f[ks][tt], acc[tt]);
                // issue group g+2 loads
#pragma unroll
                for (int ks = 0; ks < 4; ++ks) {
                    a0[ks] = *(const v2f*)(hrow_lds + k2 + 4 * ks);
#pragma unroll
                    for (int tt = 0; tt < 4; ++tt)
                        b0f[ks][tt] = *(const v2f*)(brow_wr[tt] + k2 + 4 * ks);
                }
                // consume group g+1
#pragma unroll
                for (int ks = 0; ks < 4; ++ks)
#pragma unroll
                    for (int tt = 0; tt < 4; ++tt)
                        acc[tt] = WMMA_F32(a1[ks], b1f[ks][tt], acc[tt]);
            }
        }

        __syncthreads();   // 3) all LDS reads of hl done; epilogue may overwrite

        // ---- 4) input projection: K = NINP, A from prefetched global stream ----
        {
            const float* arow = arow_in + (size_t)t * BATCH * NINP;
            v2f a0[4], a1[4], b0f[4][4], b1f[4][4];
#pragma unroll
            for (int ks = 0; ks < 4; ++ks) {
                a0[ks] = __builtin_nontemporal_load((const v2f*)(arow + 4 * ks));
#pragma unroll
                for (int tt = 0; tt < 4; ++tt)
                    b0f[ks][tt] = *(const v2f*)(brow_wi[tt] + 4 * ks);
            }
#pragma unroll 1
            for (int g = 0; g < NINP / 16; g += 2) {
                const int k1 = (g + 1) * 16;
                const int k2 = (g + 2 < NINP / 16) ? (g + 2) * 16 : 0;
#pragma unroll
                for (int ks = 0; ks < 4; ++ks) {
                    a1[ks] = __builtin_nontemporal_load((const v2f*)(arow + k1 + 4 * ks));
#pragma unroll
                    for (int tt = 0; tt < 4; ++tt)
                        b1f[ks][tt] = *(const v2f*)(brow_wi[tt] + k1 + 4 * ks);
                }
#pragma unroll
                for (int ks = 0; ks < 4; ++ks)
#pragma unroll
                    for (int tt = 0; tt < 4; ++tt)
                        acc[tt] = WMMA_F32(a0[ks], b0f[ks][tt], acc[tt]);
#pragma unroll
                for (int ks = 0; ks < 4; ++ks) {
                    a0[ks] = __builtin_nontemporal_load((const v2f*)(arow + k2 + 4 * ks));
#pragma unroll
                    for (int tt = 0; tt < 4; ++tt)
                        b0f[ks][tt] = *(const v2f*)(brow_wi[tt] + k2 + 4 * ks);
                }
#pragma unroll
                for (int ks = 0; ks < 4; ++ks)
#pragma unroll
                    for (int tt = 0; tt < 4; ++tt)
                        acc[tt] = WMMA_F32(a1[ks], b1f[ks][tt], acc[tt]);
            }
        }

        // ---- 5) noise + relu + writeback ----
        const bool last = (t == T_STEPS - 1);
        const float* nzb = noise + ((size_t)t * BATCH + b0 + 8 * lhi) * NREC;
#pragma unroll
        for (int tt = 0; tt < 4; ++tt) {
#pragma unroll
            for (int r = 0; r < 8; ++r) {
                const int m = r + 8 * lhi;
                float nz = __builtin_nontemporal_load(nzb + (size_t)r * NREC + ncol[tt]);
                float v = acc[tt][r] + nz;
                v = v > 0.0f ? v : 0.0f;
                hl[m * LDS_STRIDE + ncol[tt]] = v;
                __builtin_nontemporal_store(
                    v, &h_out[((size_t)t * BATCH + b0 + m) * NREC + ncol[tt]]);
                if (last)
                    h_final[(size_t)(b0 + m) * NREC + ncol[tt]] = v;
            }
        }
    }
}

// -----------------------------------------------------------------------------
// Output head: softmax(h_final @ Wo^T + bo).  Grid: 16 blocks x 128 threads
// (4 waves); wave w computes the 16x16 logit tile for columns [w*16, w*16+16),
// K=512 WMMA f32 loop; then a tiny LDS row-softmax.
// -----------------------------------------------------------------------------
__global__ __launch_bounds__(128) void rnn_out_kernel(
    const float* __restrict__ h_final,  // [B,NREC]
    const float* __restrict__ Wo,       // [NOUT,NREC]
    const float* __restrict__ bo,       // [NOUT]
    float* __restrict__ out)            // [B,NOUT]
{
    __shared__ float lg[16 * NOUT];

    const int tid  = threadIdx.x;
    const int wave = tid >> 5;
    const int lane = tid & 31;
    const int l16  = lane & 15;
    const int lhi  = lane >> 4;
    const int b0   = blockIdx.x * 16;
    const int nloc = wave * 16 + l16;     // 0..63

    v8f acc;
    const float bias = bo[nloc];
#pragma unroll
    for (int r = 0; r < 8; ++r) acc[r] = bias;

    const float* arow = h_final + (size_t)(b0 + l16) * NREC + 2 * lhi;
    const float* brow = Wo + (size_t)nloc * NREC + 2 * lhi;
#pragma unroll 4
    for (int k = 0; k < NREC; k += 4) {
        v2f a = *(const v2f*)(arow + k);
        v2f b = *(const v2f*)(brow + k);
        acc = WMMA_F32(a, b, acc);
    }

#pragma unroll
    for (int r = 0; r < 8; ++r)
        lg[(r + 8 * lhi) * NOUT + nloc] = acc[r];

    __syncthreads();

    if (tid < 16) {
        const int m = tid;
        float mx = -3.0e38f;
        for (int j = 0; j < NOUT; ++j) mx = fmaxf(mx, lg[m * NOUT + j]);
        float s = 0.0f;
        for (int j = 0; j < NOUT; ++j) s += __expf(lg[m * NOUT + j] - mx);
        const float inv = 1.0f / s;
        for (int j = 0; j < NOUT; ++j)
            out[(size_t)(b0 + m) * NOUT + j] = __expf(lg[m * NOUT + j] - mx) * inv;
    }
}

// -----------------------------------------------------------------------------
extern "C" void kernel_launch(void* const* d_in, const int* in_sizes, int n_in,
                              void* d_out, int out_size, void* d_ws, size_t ws_size,
                              hipStream_t stream) {
    (void)in_sizes; (void)n_in; (void)out_size; (void)d_ws; (void)ws_size;
    const float* inputs = (const float*)d_in[0];
    const float* noise  = (const float*)d_in[1];
    const float* Wi     = (const float*)d_in[2];
    const float* bi     = (const float*)d_in[3];
    const float* Wrec   = (const float*)d_in[4];
    const float* Wo     = (const float*)d_in[5];
    const float* bo     = (const float*)d_in[6];

    float* out    = (float*)d_out;                            // [B,NOUT]
    float* h      = out + (size_t)BATCH * NOUT;               // [T,B,NREC]
    float* h_fin  = h + (size_t)T_STEPS * BATCH * NREC;       // [B,NREC]

    rnn_scan_kernel<<<BATCH / 16, 256, 0, stream>>>(inputs, noise, Wi, bi, Wrec, h, h_fin);
    rnn_out_kernel<<<BATCH / 16, 128, 0, stream>>>(h_fin, Wo, bo, out);
}